// Max1_82815559401938
// MI455X (gfx1250) — compile-verified
//
#include <hip/hip_runtime.h>
#include <stdint.h>

// Problem constants (match reference setup_inputs: B=64, C=1, H=W=1024)
#define BATCH      64
#define NPB        (1u << 20)        // elements per batch sample
#define TOPN       1000u

// Level-1 histogram: bin = (|x| bits) >> 18  -> max 8159 < 8192 bins
#define NBIN       8192
#define SHIFT      18

#define CAND_MAX   4096              // candidates in the threshold bin (expect ~200)
#define SEL_MAX    1024              // selected indices per batch (exactly 1000)

// TDM streaming geometry for the histogram kernel
#define TILE            2048                 // floats per DMA (8 KB)
#define BLKS_PER_BATCH  32
#define CHUNK           (NPB / BLKS_PER_BATCH)   // 32768 elements per block
#define ITERS           (CHUNK / TILE)           // 16 DMAs per block

// Workspace layout (uint32 word offsets)
#define HIST_W    0u
#define CNT_W     (BATCH * NBIN)                 // candidate counters
#define SELCNT_W  (CNT_W + BATCH)                // selected counters
#define PRM_W     (SELCNT_W + BATCH)             // per-batch params: [0]=T1 [1]=count_above [2]=needed
#define SELIDX_W  (PRM_W + BATCH * 8)            // selected indices
#define CAND_W    (SELIDX_W + BATCH * SEL_MAX)   // uint2 {bits, idx} candidates
#define ZERO_WORDS SELIDX_W                      // hist + counters + params must be zeroed each call

typedef unsigned int v4u __attribute__((ext_vector_type(4)));
typedef unsigned int v8u __attribute__((ext_vector_type(8)));

// ---------------------------------------------------------------------------
// Tensor Data Mover: DMA `nelem` consecutive fp32 from global into LDS.
// D# per CDNA5 ISA: group0 = {count/flags, lds_addr,
// global_addr lo, global_addr hi | type=2}; group1 = {data_size=4B, dims}.
// 2-group form => VADDR2/VADDR3 NULL (tensors up to 2D). Issued once per wave.
// ---------------------------------------------------------------------------
__device__ __forceinline__ void tdm_load_f32(uint32_t lds_addr, const float* gptr,
                                             uint32_t nelem) {
  const uint64_t ga = (uint64_t)(uintptr_t)gptr;
  v4u g0;
  g0.x = 1u;                                   // count=1, is_restore=0, gather=0
  g0.y = lds_addr;                             // LDS byte address
  g0.z = (uint32_t)ga;                         // global_addr[31:0]
  g0.w = ((uint32_t)(ga >> 32) & 0x01FFFFFFu)  // global_addr[56:32]
         | 0x80000000u;                        // type=2 ("image") at bits[127:126]
  v8u g1;
  g1.s0 = (2u << 16);                          // workgroup_mask=0, data_size=2 (4B)
  g1.s1 = (nelem & 0xFFFFu) << 16;             // tensor_dim0[15:0] at bits[63:48]
  g1.s2 = (nelem >> 16) | (1u << 16);          // tensor_dim0[31:16] | tensor_dim1=1
  g1.s3 = (nelem & 0xFFFFu) << 16;             // tensor_dim1 hi=0 | tile_dim0
  g1.s4 = 1u;                                  // tile_dim1=1, tile_dim2=0
  g1.s5 = nelem;                               // tensor_dim0_stride[31:0]
  g1.s6 = 0u;                                  // stride0 hi | stride1 lo
  g1.s7 = 0u;
  asm volatile("tensor_load_to_lds %0, %1" :: "s"(g0), "s"(g1) : "memory");
}

// ---------------------------------------------------------------------------
__global__ __launch_bounds__(256) void k_zero(uint32_t* __restrict__ ws) {
  uint32_t i = blockIdx.x * blockDim.x + threadIdx.x;
  if (i < ZERO_WORDS) ws[i] = 0u;
}

// weight -> out (cold branch of the reference is exactly this; hot branch adds
// the scatter on top in k_scatter)
__global__ __launch_bounds__(256) void k_copy(const float* __restrict__ w,
                                              float* __restrict__ out,
                                              size_t nvec4) {
  size_t i = (size_t)blockIdx.x * blockDim.x + threadIdx.x;
  const size_t stride = (size_t)gridDim.x * blockDim.x;
  const float4* src = (const float4*)w;
  float4* dst = (float4*)out;
  for (; i < nvec4; i += stride) dst[i] = src[i];
}

// Per-batch 8192-bin histogram of (|x| bits)>>18. Data is streamed into LDS by
// the TDM (double-buffered, issued by wave 0), binned with LDS atomics, and
// flushed once with global atomics.
__global__ __launch_bounds__(256) void k_hist(const float* __restrict__ diff,
                                              uint32_t* __restrict__ ws) {
  __shared__ uint32_t lhist[NBIN];     // 32 KB
  __shared__ float stage[2][TILE];     // 2 x 8 KB staging

  const uint32_t b   = blockIdx.x;
  const uint32_t blk = blockIdx.y;
  const float* gsrc  = diff + (size_t)b * NPB + (size_t)blk * CHUNK;
  const bool issuer  = (threadIdx.x >> 5) == 0;   // wave 0 drives the TDM

  for (uint32_t i = threadIdx.x; i < NBIN; i += blockDim.x) lhist[i] = 0u;

  if (issuer) {
    tdm_load_f32((uint32_t)(uintptr_t)&stage[0][0], gsrc, TILE);
  }

  for (uint32_t t = 0; t < ITERS; ++t) {
    if (issuer) {
      if (t + 1 < ITERS) {
        tdm_load_f32((uint32_t)(uintptr_t)&stage[(t + 1) & 1][0],
                     gsrc + (size_t)(t + 1) * TILE, TILE);
        __builtin_amdgcn_s_wait_tensorcnt(1);   // oldest DMA (tile t) done
      } else {
        __builtin_amdgcn_s_wait_tensorcnt(0);
      }
    }
    __syncthreads();                            // tile t visible to all waves
    const float4* sv = (const float4*)&stage[t & 1][0];
    for (uint32_t i = threadIdx.x; i < TILE / 4; i += blockDim.x) {
      const float4 v = sv[i];
      atomicAdd(&lhist[(__float_as_uint(v.x) & 0x7fffffffu) >> SHIFT], 1u);
      atomicAdd(&lhist[(__float_as_uint(v.y) & 0x7fffffffu) >> SHIFT], 1u);
      atomicAdd(&lhist[(__float_as_uint(v.z) & 0x7fffffffu) >> SHIFT], 1u);
      atomicAdd(&lhist[(__float_as_uint(v.w) & 0x7fffffffu) >> SHIFT], 1u);
    }
    __syncthreads();                            // buffer reusable by next DMA
  }

  uint32_t* gh = ws + HIST_W + (size_t)b * NBIN;
  for (uint32_t i = threadIdx.x; i < NBIN; i += blockDim.x) {
    const uint32_t v = lhist[i];
    if (v) atomicAdd(&gh[i], v);
  }
}

// Per batch: find threshold bin T1 with cum(T1+1) < TOPN <= cum(T1)
__global__ __launch_bounds__(256) void k_sel1(uint32_t* __restrict__ ws) {
  __shared__ uint32_t ps[256];
  const uint32_t b = blockIdx.x;
  const uint32_t* h = ws + HIST_W + (size_t)b * NBIN;
  uint32_t s = 0;
  const uint32_t base = threadIdx.x * (NBIN / 256);
  for (uint32_t i = 0; i < NBIN / 256; ++i) s += h[base + i];
  ps[threadIdx.x] = s;
  __syncthreads();
  if (threadIdx.x == 0) {
    uint32_t acc = 0;
    int seg = 255;
    for (; seg > 0; --seg) {
      if (acc + ps[seg] >= TOPN) break;
      acc += ps[seg];
    }
    uint32_t T1 = 0;
    for (int bin = seg * (NBIN / 256) + (NBIN / 256) - 1;
         bin >= seg * (NBIN / 256); --bin) {
      const uint32_t v = h[bin];
      if (acc + v >= TOPN) { T1 = (uint32_t)bin; break; }
      acc += v;
    }
    ws[PRM_W + b * 8 + 0] = T1;
    ws[PRM_W + b * 8 + 1] = acc;          // count strictly above bin T1
    ws[PRM_W + b * 8 + 2] = TOPN - acc;   // needed from bin T1
  }
}

// bins > T1 -> selected indices directly; bin == T1 -> candidate list
__global__ __launch_bounds__(256) void k_collect(const float* __restrict__ diff,
                                                 uint32_t* __restrict__ ws) {
  const uint32_t b  = blockIdx.y;
  const uint32_t T1 = ws[PRM_W + b * 8 + 0];
  const uint32_t e0 = (blockIdx.x * blockDim.x + threadIdx.x) * 4;
  const float4 d = *(const float4*)(diff + (size_t)b * NPB + e0);
  uint32_t bv[4] = {__float_as_uint(d.x), __float_as_uint(d.y),
                    __float_as_uint(d.z), __float_as_uint(d.w)};
#pragma unroll
  for (int c = 0; c < 4; ++c) {
    const uint32_t bits = bv[c] & 0x7fffffffu;
    const uint32_t bin  = bits >> SHIFT;
    if (bin > T1) {
      const uint32_t p = atomicAdd(&ws[SELCNT_W + b], 1u);
      if (p < SEL_MAX) ws[SELIDX_W + b * SEL_MAX + p] = e0 + (uint32_t)c;
    } else if (bin == T1) {
      const uint32_t p = atomicAdd(&ws[CNT_W + b], 1u);
      if (p < CAND_MAX) {
        uint2* cand = (uint2*)(ws + CAND_W);
        cand[(size_t)b * CAND_MAX + p] = make_uint2(bits, e0 + (uint32_t)c);
      }
    }
  }
}

// Exact rank among threshold-bin candidates (ties broken by smaller index,
// matching jax.lax.top_k stability); append winners to the selected list.
__global__ __launch_bounds__(256) void k_sel2(uint32_t* __restrict__ ws) {
  __shared__ uint2 lc[CAND_MAX];   // 32 KB
  const uint32_t b = blockIdx.x;
  uint32_t n = ws[CNT_W + b];
  if (n > CAND_MAX) n = CAND_MAX;
  const uint32_t count_above = ws[PRM_W + b * 8 + 1];
  const uint2* cand = (const uint2*)(ws + CAND_W) + (size_t)b * CAND_MAX;
  for (uint32_t i = threadIdx.x; i < n; i += blockDim.x) lc[i] = cand[i];
  __syncthreads();
  for (uint32_t i = threadIdx.x; i < n; i += blockDim.x) {
    const uint32_t bits = lc[i].x, idx = lc[i].y;
    uint32_t r = count_above;
    for (uint32_t j = 0; j < n; ++j) {
      const uint2 cj = lc[j];
      r += (uint32_t)((cj.x > bits) || (cj.x == bits && cj.y < idx));
    }
    if (r < TOPN) {
      const uint32_t p = atomicAdd(&ws[SELCNT_W + b], 1u);
      if (p < SEL_MAX) ws[SELIDX_W + b * SEL_MAX + p] = idx;
    }
  }
}

// Hot-branch scatter: +1.0 at exactly the TOPN selected positions per batch.
// Epoch gate evaluated on-device so the launch sequence is capture-safe.
__global__ __launch_bounds__(256) void k_scatter(const uint32_t* __restrict__ ws,
                                                 const int* __restrict__ epoch,
                                                 float* __restrict__ out) {
  const int e = *epoch;
  if (!((e > 1000) && (e < 18000) && (e % 200 == 0))) return;
  const uint32_t b = blockIdx.x;
  uint32_t n = ws[SELCNT_W + b];
  if (n > SEL_MAX) n = SEL_MAX;
  float* o = out + (size_t)b * NPB;
  for (uint32_t j = threadIdx.x; j < n; j += blockDim.x) {
    const uint32_t idx = ws[SELIDX_W + b * SEL_MAX + j];
    o[idx] += 1.0f;   // indices are unique -> plain RMW is safe
  }
}

// ---------------------------------------------------------------------------
extern "C" void kernel_launch(void* const* d_in, const int* in_sizes, int n_in,
                              void* d_out, int out_size, void* d_ws, size_t ws_size,
                              hipStream_t stream) {
  (void)in_sizes; (void)n_in; (void)out_size; (void)ws_size;
  const float* diff   = (const float*)d_in[0];
  const float* weight = (const float*)d_in[1];
  const int*   epoch  = (const int*)d_in[2];
  float*       out    = (float*)d_out;
  uint32_t*    ws     = (uint32_t*)d_ws;

  k_zero<<<(ZERO_WORDS + 255) / 256, 256, 0, stream>>>(ws);
  k_copy<<<8192, 256, 0, stream>>>(weight, out, (size_t)BATCH * NPB / 4);
  k_hist<<<dim3(BATCH, BLKS_PER_BATCH), 256, 0, stream>>>(diff, ws);
  k_sel1<<<BATCH, 256, 0, stream>>>(ws);
  k_collect<<<dim3(NPB / (4 * 256), BATCH), 256, 0, stream>>>(diff, ws);
  k_sel2<<<BATCH, 256, 0, stream>>>(ws);
  k_scatter<<<BATCH, 256, 0, stream>>>(ws, epoch, out);
}